// LadanPPK_Criminal_49624052138312
// MI455X (gfx1250) — compile-verified
//
#include <hip/hip_runtime.h>

// ---------------------------------------------------------------------------
// Fused segment max/min/mean for MI455X (gfx1250, wave32).
//
// features: [N, 256] f32, group_index: [N] i32 in [0,64). Output [64, 768]:
// per group [max(256) || min(256) || mean(256)].
//
// Bandwidth-bound: 1 GiB stream / 23.3 TB/s ~= 46 us floor.
// - max/min: LDS-privatized u32-key atomics (2 ds atomics / element).
// - sum:     exact f32 one-hot GEMM on the matrix pipe via
//            V_WMMA_F32_16X16X4_F32 (M=16 groups x N=16 cols x K=4 rows),
//            4 accumulators per wave cover all 64 groups.
// ---------------------------------------------------------------------------

typedef __attribute__((ext_vector_type(2))) float v2f;
typedef __attribute__((ext_vector_type(8))) float v8f;

#define GROUPS 64

// Monotone float -> u32 key: order-preserving for all finite values.
__device__ __forceinline__ unsigned encodeKey(float f) {
  unsigned u = __float_as_uint(f);
  return (u & 0x80000000u) ? ~u : (u | 0x80000000u);
}
__device__ __forceinline__ float decodeKey(unsigned k) {
  unsigned u = (k & 0x80000000u) ? (k & 0x7FFFFFFFu) : ~k;
  return __uint_as_float(u);
}

// ---------------------------------------------------------------------------
// Kernel 0: initialize workspace accumulators (d_ws is poisoned, and we must
// be replay-safe, so re-init every call).
// ---------------------------------------------------------------------------
__global__ void init_ws_kernel(unsigned* __restrict__ gmax,
                               unsigned* __restrict__ gmin,
                               float* __restrict__ gsum,
                               unsigned* __restrict__ gcount, int gd) {
  int i = blockIdx.x * blockDim.x + threadIdx.x;
  if (i < gd) {
    gmax[i] = 0u;           // key(-inf) = 0x007FFFFF > 0, so 0 is a safe -identity
    gmin[i] = 0xFFFFFFFFu;  // +identity for min over keys
    gsum[i] = 0.0f;
  }
  if (i < GROUPS) gcount[i] = 0u;
}

// ---------------------------------------------------------------------------
// Kernel 1: streaming segment reduction.
// grid.x = D/16 column tiles, grid.y = row chunks. 256 threads = 8 waves.
// Each wave steps 4 rows x 16 cols per iteration (stride 32 rows in block).
//
// Per-lane operand mapping for V_WMMA_F32_16X16X4_F32 (wave32):
//   A (16x4, one-hot groups): lane L -> M = L&15; v0 holds K = (L<16?0:2),
//                                          v1 holds K = (L<16?1:3)
//   B (4x16, features):       lane L -> N = L&15; same K mapping as A
//   C (16x16 f32):            lane L -> N = L&15; VGPR j -> M = j + (L<16?0:8)
// So each lane loads feat[row+k][colbase + (L&15)] for its two K values — the
// exact same two values feed the LDS max/min atomics. No data is touched twice.
// ---------------------------------------------------------------------------
__global__ __launch_bounds__(256)
void seg_reduce_wmma_kernel(const float* __restrict__ feat,
                            const int* __restrict__ gidx,
                            unsigned* __restrict__ gmax,
                            unsigned* __restrict__ gmin,
                            float* __restrict__ gsum,
                            unsigned* __restrict__ gcount,
                            int Ntot, int Dtot, int rowsPerBlock) {
  __shared__ unsigned smax[GROUPS * 16];  // [group][col-in-tile] keys
  __shared__ unsigned smin[GROUPS * 16];
  __shared__ unsigned scount[GROUPS];

  const int tid = threadIdx.x;
  for (int i = tid; i < GROUPS * 16; i += 256) {
    smax[i] = 0u;
    smin[i] = 0xFFFFFFFFu;
  }
  if (tid < GROUPS) scount[tid] = 0u;
  __syncthreads();

  const int lane = tid & 31;
  const int wave = tid >> 5;
  const int n = lane & 15;              // column within tile == WMMA N == WMMA M
  const int khalf = (lane >> 4) << 1;   // 0 (lanes 0-15) or 2 (lanes 16-31)
  const int colbase = blockIdx.x * 16;
  const bool doCount = (blockIdx.x == 0) && (n == 0);  // lanes 0 & 16 only

  v8f c0 = {}, c1 = {}, c2 = {}, c3 = {};  // 64 groups x 16 cols of sums

  long rowStart = (long)blockIdx.y * rowsPerBlock + (long)wave * 4;
  long rowEnd = (long)(blockIdx.y + 1) * rowsPerBlock;
  if (rowEnd > Ntot) rowEnd = Ntot;

  for (long row = rowStart; row + 4 <= rowEnd; row += 32) {
    // groups of the 4 rows in this chunk (row is 4-aligned -> int4 load OK)
    const int4 g4 = *(const int4*)(gidx + row);
    const int g0 = (khalf == 0) ? g4.x : g4.z;  // group of row+khalf
    const int g1 = (khalf == 0) ? g4.y : g4.w;  // group of row+khalf+1

    // B-matrix / max-min feature values (coalesced 64B row segments)
    const float f0 = feat[(row + khalf) * (long)Dtot + colbase + n];
    const float f1 = feat[(row + khalf + 1) * (long)Dtot + colbase + n];

    // ---- max/min via LDS u32 atomics (2 atomics per element) ----
    const unsigned k0e = encodeKey(f0);
    const unsigned k1e = encodeKey(f1);
    atomicMax(&smax[(g0 << 4) + n], k0e);
    atomicMin(&smin[(g0 << 4) + n], k0e);
    atomicMax(&smax[(g1 << 4) + n], k1e);
    atomicMin(&smin[(g1 << 4) + n], k1e);

    // ---- row counts (column-tile 0 only; each row counted exactly once) ----
    if (doCount) {
      atomicAdd(&scount[g0], 1u);
      atomicAdd(&scount[g1], 1u);
    }

    // ---- sums via exact f32 one-hot WMMA ----
    v2f b;
    b.x = f0;
    b.y = f1;
    v2f a;
    a.x = (g0 == n) ? 1.0f : 0.0f;
    a.y = (g1 == n) ? 1.0f : 0.0f;
    c0 = __builtin_amdgcn_wmma_f32_16x16x4_f32(false, a, false, b, (short)0,
                                               c0, false, false);
    a.x = (g0 == 16 + n) ? 1.0f : 0.0f;
    a.y = (g1 == 16 + n) ? 1.0f : 0.0f;
    c1 = __builtin_amdgcn_wmma_f32_16x16x4_f32(false, a, false, b, (short)0,
                                               c1, false, false);
    a.x = (g0 == 32 + n) ? 1.0f : 0.0f;
    a.y = (g1 == 32 + n) ? 1.0f : 0.0f;
    c2 = __builtin_amdgcn_wmma_f32_16x16x4_f32(false, a, false, b, (short)0,
                                               c2, false, false);
    a.x = (g0 == 48 + n) ? 1.0f : 0.0f;
    a.y = (g1 == 48 + n) ? 1.0f : 0.0f;
    c3 = __builtin_amdgcn_wmma_f32_16x16x4_f32(false, a, false, b, (short)0,
                                               c3, false, false);
  }

  // ---- merge this wave's partial sums into global workspace ----
  const int mrow = (lane >> 4) << 3;  // 0 or 8
#pragma unroll
  for (int j = 0; j < 8; ++j) {
    const int m = j + mrow;
    unsafeAtomicAdd(&gsum[(m)*Dtot + colbase + n], c0[j]);
    unsafeAtomicAdd(&gsum[(16 + m) * Dtot + colbase + n], c1[j]);
    unsafeAtomicAdd(&gsum[(32 + m) * Dtot + colbase + n], c2[j]);
    unsafeAtomicAdd(&gsum[(48 + m) * Dtot + colbase + n], c3[j]);
  }

  // ---- merge LDS max/min tiles into global workspace ----
  __syncthreads();
  for (int i = tid; i < GROUPS * 16; i += 256) {
    const int g = i >> 4;
    const int cc = i & 15;
    atomicMax(&gmax[g * Dtot + colbase + cc], smax[i]);
    atomicMin(&gmin[g * Dtot + colbase + cc], smin[i]);
  }
  if (blockIdx.x == 0 && tid < GROUPS) atomicAdd(&gcount[tid], scount[tid]);
}

// ---------------------------------------------------------------------------
// Kernel 2: finalize -> out[g][0:D]=max, [D:2D]=min, [2D:3D]=sum/count
// ---------------------------------------------------------------------------
__global__ void finalize_kernel(const unsigned* __restrict__ gmax,
                                const unsigned* __restrict__ gmin,
                                const float* __restrict__ gsum,
                                const unsigned* __restrict__ gcount,
                                float* __restrict__ out, int Dtot) {
  const int i = blockIdx.x * blockDim.x + threadIdx.x;
  const int gd = GROUPS * Dtot;
  if (i >= gd) return;
  const int g = i / Dtot;
  const int cc = i - g * Dtot;
  const float cnt = (float)gcount[g];
  float* o = out + (long)g * (3 * Dtot);
  o[cc] = decodeKey(gmax[i]);
  o[Dtot + cc] = decodeKey(gmin[i]);
  o[2 * Dtot + cc] = gsum[i] / cnt;
}

// ---------------------------------------------------------------------------
extern "C" void kernel_launch(void* const* d_in, const int* in_sizes, int n_in,
                              void* d_out, int out_size, void* d_ws,
                              size_t ws_size, hipStream_t stream) {
  const float* feat = (const float*)d_in[0];
  const int* gidx = (const int*)d_in[1];
  float* out = (float*)d_out;

  const int Ntot = in_sizes[1];           // 1048576 (multiple of 32)
  const int Dtot = in_sizes[0] / Ntot;    // 256 (multiple of 16)
  const int gd = GROUPS * Dtot;           // 16384

  // Workspace layout (~197 KB): max keys | min keys | sums | counts
  unsigned* gmax = (unsigned*)d_ws;
  unsigned* gmin = gmax + gd;
  float* gsum = (float*)(gmin + gd);
  unsigned* gcount = (unsigned*)(gsum + gd);

  init_ws_kernel<<<(gd + 255) / 256, 256, 0, stream>>>(gmax, gmin, gsum,
                                                       gcount, gd);

  // 8192 rows per block (multiple of 32) -> 128 row chunks x 16 col tiles
  // = 2048 blocks of 8 waves; ~21M global merge atomics total (negligible).
  const int rowsPerBlock = 8192;
  const int yb = (Ntot + rowsPerBlock - 1) / rowsPerBlock;
  dim3 grid(Dtot / 16, yb);
  seg_reduce_wmma_kernel<<<grid, 256, 0, stream>>>(
      feat, gidx, gmax, gmin, gsum, gcount, Ntot, Dtot, rowsPerBlock);

  finalize_kernel<<<(gd + 255) / 256, 256, 0, stream>>>(gmax, gmin, gsum,
                                                        gcount, out, Dtot);
}